// head_attention_vanilla_19645180412018
// MI455X (gfx1250) — compile-verified
//
#include <hip/hip_runtime.h>

typedef __attribute__((ext_vector_type(2))) float v2f;
typedef __attribute__((ext_vector_type(8))) float v8f;

#define NB 4
#define NN 512
#define ND 64
#define NH 64

// ---------------------------------------------------------------------------
// Kernel 1: node projections  Q,K,V,Ni,Nj = x @ W + b   ([2048,64]x[64,64])
// Trivial FLOP count (84 MFLOP total) -> simple VALU kernel, coalesced W reads.
// ---------------------------------------------------------------------------
__global__ __launch_bounds__(256) void node_proj_kernel(
    const float* __restrict__ x,
    const float* __restrict__ Wq, const float* __restrict__ bq,
    const float* __restrict__ Wk, const float* __restrict__ bk,
    const float* __restrict__ Wv, const float* __restrict__ bv,
    const float* __restrict__ Wni, const float* __restrict__ bni,
    const float* __restrict__ Wnj, const float* __restrict__ bnj,
    float* __restrict__ Q, float* __restrict__ K, float* __restrict__ V,
    float* __restrict__ Ni, float* __restrict__ Nj)
{
    int idx   = blockIdx.x * 256 + threadIdx.x;      // 0 .. 5*2048*64-1
    int which = idx >> 17;                           // 2048*64 = 1<<17
    int rem   = idx & ((1 << 17) - 1);
    int row   = rem >> 6;
    int h     = rem & 63;

    const float* W; const float* bb; float* out;
    switch (which) {
        case 0:  W = Wq;  bb = bq;  out = Q;  break;
        case 1:  W = Wk;  bb = bk;  out = K;  break;
        case 2:  W = Wv;  bb = bv;  out = V;  break;
        case 3:  W = Wni; bb = bni; out = Ni; break;
        default: W = Wnj; bb = bnj; out = Nj; break;
    }

    const float* xr = x + row * ND;
    float acc = bb[h];
#pragma unroll
    for (int d = 0; d < ND; ++d)
        acc = fmaf(xr[d], W[d * NH + h], acc);
    out[rem] = acc;
}

// ---------------------------------------------------------------------------
// Kernel 2: the big fused pass (single read of e, single write of e_new):
//   e_new[b,i,j,h] = (e @ We)[...] + be[h] + Ni[b,i,h] + Nj[b,j,h]
//   att[b,i,j]     = (1/8) * sum_h Q[b,i,h] * e_new[b,i,j,h] * K[b,j,h]
// One wave handles a 16(j) x 64(h) tile via 64x v_wmma_f32_16x16x4_f32.
// We is staged transposed in LDS with pitch 68 -> 8B-aligned ds_load_b64,
// bank = (4n+k) mod 64 hits all 64 banks exactly once (conflict-free).
// ---------------------------------------------------------------------------
__global__ __launch_bounds__(256) void edge_wmma_kernel(
    const float* __restrict__ e,  const float* __restrict__ We,
    const float* __restrict__ be, const float* __restrict__ Q,
    const float* __restrict__ K,  const float* __restrict__ Ni,
    const float* __restrict__ Nj,
    float* __restrict__ e_new, float* __restrict__ att)
{
    __shared__ float weT[64 * 68];                   // We transposed, pitch 68

    int tid = threadIdx.x;
#pragma unroll
    for (int s = 0; s < 16; ++s) {                   // 256 thr * 16 = 4096 elems
        int idx = tid + 256 * s;
        int k = idx >> 6, n = idx & 63;
        weT[n * 68 + k] = We[idx];                   // weT[n][k] = We[k][n]
    }
    __syncthreads();

    int lane  = tid & 31;
    int wave  = tid >> 5;
    int tile  = blockIdx.x * 8 + wave;               // 0 .. 65535
    int jblk  = tile & 31;
    int i     = (tile >> 5) & (NN - 1);
    int b     = tile >> 14;
    int j0    = jblk << 4;

    int nlo   = lane & 15;                           // N-col / M-row within half
    int khi   = (lane >> 4) << 1;                    // K offset: 0 or 2 (ISA A layout)
    int half8 = (lane >> 4) << 3;                    // D-matrix M split: 0 or 8

    // A-tile: 16 consecutive j-rows of e (contiguous 4KB block)
    const float* eA =
        e + (((size_t)(b * NN + i)) * NN + (j0 + nlo)) * (size_t)ND + khi;

    v8f acc0 = {}, acc1 = {}, acc2 = {}, acc3 = {};
#pragma unroll
    for (int c = 0; c < 16; ++c) {                   // K = 64 in chunks of 4
        v2f af = *(const v2f*)(eA + 4 * c);
        int k = 4 * c + khi;
        v2f bf0 = *(const v2f*)(&weT[( 0 + nlo) * 68 + k]);
        v2f bf1 = *(const v2f*)(&weT[(16 + nlo) * 68 + k]);
        v2f bf2 = *(const v2f*)(&weT[(32 + nlo) * 68 + k]);
        v2f bf3 = *(const v2f*)(&weT[(48 + nlo) * 68 + k]);
        acc0 = __builtin_amdgcn_wmma_f32_16x16x4_f32(false, af, false, bf0, (short)0, acc0, false, false);
        acc1 = __builtin_amdgcn_wmma_f32_16x16x4_f32(false, af, false, bf1, (short)0, acc1, false, false);
        acc2 = __builtin_amdgcn_wmma_f32_16x16x4_f32(false, af, false, bf2, (short)0, acc2, false, false);
        acc3 = __builtin_amdgcn_wmma_f32_16x16x4_f32(false, af, false, bf3, (short)0, acc3, false, false);
    }

    // Epilogue: bias + Ni + Nj, store e_new, fused logit reduction over h
    size_t rowI = (size_t)(b * NN + i) * NH;
    const float* Qi  = Q  + rowI;
    const float* Nii = Ni + rowI;
    size_t ebase = ((size_t)(b * NN + i)) * NN * (size_t)NH;

    float sarr[8];
#pragma unroll
    for (int r = 0; r < 8; ++r) {
        int jl = r + half8;                          // D layout: VGPR r -> M=r / r+8
        int jg = j0 + jl;
        size_t rowJ = (size_t)(b * NN + jg) * NH;
        float s = 0.f;
#pragma unroll
        for (int t = 0; t < 4; ++t) {
            int h = t * 16 + nlo;
            float a = (t == 0) ? acc0[r] : (t == 1) ? acc1[r]
                    : (t == 2) ? acc2[r] : acc3[r];
            float val = a + be[h] + Nii[h] + Nj[rowJ + h];
            e_new[ebase + (size_t)jg * NH + h] = val;
            s = fmaf(Qi[h] * K[rowJ + h], val, s);
        }
        // butterfly sum over the 16 lanes holding this j-row's 64 h-columns
        s += __shfl_xor(s, 1);
        s += __shfl_xor(s, 2);
        s += __shfl_xor(s, 4);
        s += __shfl_xor(s, 8);
        sarr[r] = s;
    }

    float myv = sarr[0];
    int rsel = nlo & 7;
    if (rsel == 1) myv = sarr[1];
    if (rsel == 2) myv = sarr[2];
    if (rsel == 3) myv = sarr[3];
    if (rsel == 4) myv = sarr[4];
    if (rsel == 5) myv = sarr[5];
    if (rsel == 6) myv = sarr[6];
    if (rsel == 7) myv = sarr[7];
    if (nlo < 8)
        att[(size_t)(b * NN + i) * NN + j0 + half8 + nlo] = myv * 0.125f;
}

// ---------------------------------------------------------------------------
// Kernel 3: softmax over dim=1 (over i, for each (b,j) column). 4MB, L2 hot.
// ---------------------------------------------------------------------------
__global__ __launch_bounds__(128) void softmax_dim1_kernel(float* __restrict__ att)
{
    __shared__ float red[128];
    int b   = blockIdx.x >> 9;
    int j   = blockIdx.x & (NN - 1);
    int tid = threadIdx.x;
    size_t base = (size_t)b * NN * NN + j;

    float v0 = att[base + (size_t)(tid       ) * NN];
    float v1 = att[base + (size_t)(tid + 128 ) * NN];
    float v2 = att[base + (size_t)(tid + 256 ) * NN];
    float v3 = att[base + (size_t)(tid + 384 ) * NN];

    float m = fmaxf(fmaxf(v0, v1), fmaxf(v2, v3));
    red[tid] = m;
    __syncthreads();
    for (int s = 64; s > 0; s >>= 1) {
        if (tid < s) red[tid] = fmaxf(red[tid], red[tid + s]);
        __syncthreads();
    }
    m = red[0];
    __syncthreads();

    v0 = __expf(v0 - m); v1 = __expf(v1 - m);
    v2 = __expf(v2 - m); v3 = __expf(v3 - m);
    red[tid] = v0 + v1 + v2 + v3;
    __syncthreads();
    for (int s = 64; s > 0; s >>= 1) {
        if (tid < s) red[tid] += red[tid + s];
        __syncthreads();
    }
    float inv = 1.0f / red[0];

    att[base + (size_t)(tid       ) * NN] = v0 * inv;
    att[base + (size_t)(tid + 128 ) * NN] = v1 * inv;
    att[base + (size_t)(tid + 256 ) * NN] = v2 * inv;
    att[base + (size_t)(tid + 384 ) * NN] = v3 * inv;
}

// ---------------------------------------------------------------------------
// Kernel 4: out[b,i,h] = sum_j att[b,i,j] * V[b,j,h]   (0.13 GFLOP, trivial)
// ---------------------------------------------------------------------------
__global__ __launch_bounds__(64) void av_kernel(
    const float* __restrict__ att, const float* __restrict__ V,
    float* __restrict__ out)
{
    int b = blockIdx.x >> 9;
    int i = blockIdx.x & (NN - 1);
    int h = threadIdx.x;
    const float* arow = att + (size_t)b * NN * NN + (size_t)i * NN;
    const float* vb   = V   + (size_t)b * NN * NH;
    float acc = 0.f;
#pragma unroll 8
    for (int j = 0; j < NN; ++j)
        acc = fmaf(arow[j], vb[j * NH + h], acc);
    out[((size_t)b * NN + i) * NH + h] = acc;
}

// ---------------------------------------------------------------------------
extern "C" void kernel_launch(void* const* d_in, const int* in_sizes, int n_in,
                              void* d_out, int out_size, void* d_ws, size_t ws_size,
                              hipStream_t stream)
{
    const float* x   = (const float*)d_in[0];
    const float* e   = (const float*)d_in[1];
    const float* Wq  = (const float*)d_in[2];  const float* bq  = (const float*)d_in[3];
    const float* Wk  = (const float*)d_in[4];  const float* bk  = (const float*)d_in[5];
    const float* We  = (const float*)d_in[6];  const float* be  = (const float*)d_in[7];
    const float* Wv  = (const float*)d_in[8];  const float* bv  = (const float*)d_in[9];
    const float* Wni = (const float*)d_in[10]; const float* bni = (const float*)d_in[11];
    const float* Wnj = (const float*)d_in[12]; const float* bnj = (const float*)d_in[13];

    float* out   = (float*)d_out;                       // [B,N,H] = 131072 floats
    float* e_new = out + (size_t)NB * NN * NH;          // [B,N,N,H] follows

    float* ws  = (float*)d_ws;
    float* Q   = ws;
    float* K   = ws + 131072;
    float* V   = ws + 262144;
    float* Ni  = ws + 393216;
    float* Nj  = ws + 524288;
    float* att = ws + 655360;                           // [B,N,N] = 1048576 floats

    node_proj_kernel<<<2560, 256, 0, stream>>>(x, Wq, bq, Wk, bk, Wv, bv,
                                               Wni, bni, Wnj, bnj,
                                               Q, K, V, Ni, Nj);
    edge_wmma_kernel<<<8192, 256, 0, stream>>>(e, We, be, Q, K, Ni, Nj,
                                               e_new, att);
    softmax_dim1_kernel<<<2048, 128, 0, stream>>>(att);
    av_kernel<<<2048, 64, 0, stream>>>(att, V, out);
}